// ScoringFunction_5308579578119
// MI455X (gfx1250) — compile-verified
//
#include <hip/hip_runtime.h>

// Problem constants (match reference)
constexpr int B_ = 4;
constexpr int N_ = 200000;
constexpr int D_ = 128;
constexpr int R_ = 200;
constexpr int E_ = 4096;

typedef __attribute__((ext_vector_type(2))) float v2f;
typedef __attribute__((ext_vector_type(8))) float v8f;

#define BDIM 256
#define SCAN_WIN 1024
#define OPAD 132   // LDS row stride (floats): 128 + 4 -> bank-conflict-free

__global__ void zero_out_kernel(float* __restrict__ out, int n) {
    int i = blockIdx.x * blockDim.x + threadIdx.x;
    if (i < n) out[i] = 0.0f;
}

// One workgroup per (relation r, batch b). 256 threads = 8 waves (wave32).
// Wave w owns output rows [16w, 16w+15] of Wo; its 16 W-rows live in registers.
__global__ __launch_bounds__(BDIM)
void score_kernel(const float* __restrict__ node_emb,   // [B][N][D] f32
                  const float* __restrict__ relW,       // [R][D][D] f32
                  const int*   __restrict__ edges,      // [B][E][3] int
                  float* __restrict__ out)              // [B][E]    f32
{
    const int r    = blockIdx.x;        // relation
    const int b    = blockIdx.y;        // batch
    const int tid  = threadIdx.x;
    const int wave = tid >> 5;          // 0..7
    const int lane = tid & 31;
    const int es   = lane & 15;         // edge slot / M-row within tile
    const bool hi  = lane >= 16;
    const int koff = hi ? 2 : 0;

    __shared__ alignas(16) float O_lds[16][OPAD];   // o-embeddings, [edge][k]
    __shared__ alignas(16) float S_lds[16][OPAD];   // s-embeddings, [edge][k]
    __shared__ float part[8][16];                   // per-wave partial scores
    __shared__ int   list[SCAN_WIN];
    __shared__ int   s_count;

    // --- Register-resident A operands: this wave's 16 rows of W, all 32 k-tiles.
    // A layout for V_WMMA_F32_16X16X4_F32: lanes 0-15 hold K={k,k+1},
    // lanes 16-31 hold K={k+2,k+3}, M = lane%16.  One b64 load per k-tile.
    v2f Areg[32];
    {
        const float* wrow = relW + (size_t)r * D_ * D_ + (size_t)(16 * wave + es) * D_;
        #pragma unroll
        for (int kt = 0; kt < 32; ++kt)
            Areg[kt] = *(const v2f*)(wrow + kt * 4 + koff);
    }

    // --- Scan this batch's edge list in windows, collecting edges of relation r.
    for (int base = 0; base < E_; base += SCAN_WIN) {
        if (tid == 0) s_count = 0;
        __syncthreads();
        int wend = base + SCAN_WIN; if (wend > E_) wend = E_;
        for (int e = base + tid; e < wend; e += BDIM) {
            int rr = edges[((size_t)b * E_ + e) * 3 + 1];
            int rc = rr < 0 ? 0 : rr;                 // clip low side
            if (rr < R_ && rc == r) {                 // rr>=R_ -> stays zero-filled
                int p = atomicAdd(&s_count, 1);
                list[p] = e;
            }
        }
        __syncthreads();
        const int cnt = s_count;

        for (int g = 0; g < cnt; g += 16) {
            int gc = cnt - g; if (gc > 16) gc = 16;   // valid edges this group

            // --- Stage 16 s/o rows into LDS (coalesced: 16 threads per row).
            {
                int slot = tid >> 4;                          // 0..15
                int f0   = (tid & 15) * 8;                    // 0..120
                int li   = g + (slot < gc ? slot : gc - 1);   // pad with last valid
                const int* ee = edges + ((size_t)b * E_ + list[li]) * 3;
                int si = ee[0]; si = si < 0 ? 0 : (si >= N_ ? N_ - 1 : si);
                int oi = ee[2]; oi = oi < 0 ? 0 : (oi >= N_ ? N_ - 1 : oi);
                const float4* srow = (const float4*)(node_emb + ((size_t)b * N_ + si) * D_ + f0);
                const float4* orow = (const float4*)(node_emb + ((size_t)b * N_ + oi) * D_ + f0);
                float4 s0 = srow[0], s1 = srow[1];
                float4 o0 = orow[0], o1 = orow[1];
                *(float4*)&S_lds[slot][f0]     = s0;
                *(float4*)&S_lds[slot][f0 + 4] = s1;
                *(float4*)&O_lds[slot][f0]     = o0;
                *(float4*)&O_lds[slot][f0 + 4] = o1;
            }
            __syncthreads();

            // --- Wo tile = sum_k A(16x4) x B(4x16), f32 WMMA, full EXEC.
            // B layout: lanes 0-15 hold rows K={k,k+1} at N=lane, lanes 16-31
            // rows K={k+2,k+3}: one b64 LDS load per k-tile.
            v8f acc = {};
            #pragma unroll
            for (int kt = 0; kt < 32; ++kt) {
                v2f bmat = *(const v2f*)&O_lds[es][kt * 4 + koff];
                acc = __builtin_amdgcn_wmma_f32_16x16x4_f32(
                          false, Areg[kt], false, bmat,
                          (short)0, acc, false, false);
            }

            // --- score contribution: dot acc with s rows (D layout:
            // VGPR j -> row 16*wave + j (+8 for hi half), col = edge slot).
            float partial = 0.0f;
            const int mbase = 16 * wave + (hi ? 8 : 0);
            #pragma unroll
            for (int j = 0; j < 8; ++j)
                partial += acc[j] * S_lds[es][mbase + j];
            partial += __shfl_xor(partial, 16, 32);   // combine row halves
            if (!hi) part[wave][es] = partial;
            __syncthreads();

            if (tid < gc) {                            // tid in [0,16)
                float sc = 0.0f;
                #pragma unroll
                for (int w = 0; w < 8; ++w) sc += part[w][tid];
                out[(size_t)b * E_ + list[g + tid]] = sc;
            }
            __syncthreads();
        }
        __syncthreads();
    }
}

extern "C" void kernel_launch(void* const* d_in, const int* in_sizes, int n_in,
                              void* d_out, int out_size, void* d_ws, size_t ws_size,
                              hipStream_t stream) {
    (void)in_sizes; (void)n_in; (void)d_ws; (void)ws_size;
    const float* node_emb = (const float*)d_in[0];
    const float* relW     = (const float*)d_in[1];
    const int*   edges    = (const int*)d_in[2];
    float*       out      = (float*)d_out;

    // Invalid relations (r >= R) must output 0: zero-fill first.
    int n = B_ * E_;
    zero_out_kernel<<<(n + 255) / 256, 256, 0, stream>>>(out, n);

    dim3 grid(R_, B_);
    score_kernel<<<grid, BDIM, 0, stream>>>(node_emb, relW, edges, out);
}